// InertiaOfBeliefMass_33363305955948
// MI455X (gfx1250) — compile-verified
//
#include <hip/hip_runtime.h>
#include <hip/hip_bf16.h>
#include <math.h>

// ---------- CDNA5 WMMA types ----------
typedef __attribute__((ext_vector_type(2))) float v2f;
typedef __attribute__((ext_vector_type(8))) float v8f;

#define EPS_ 1e-6f
#define MINEIG_ 1e-4f

// ---------- small 3x3 helpers ----------
__device__ __forceinline__ void mm3(const float* A, const float* B, float* C) {
#pragma unroll
  for (int r = 0; r < 3; ++r)
#pragma unroll
    for (int c = 0; c < 3; ++c)
      C[r*3+c] = A[r*3+0]*B[0*3+c] + A[r*3+1]*B[1*3+c] + A[r*3+2]*B[2*3+c];
}
__device__ __forceinline__ void mm3_At(const float* A, const float* B, float* C) { // A^T * B
#pragma unroll
  for (int r = 0; r < 3; ++r)
#pragma unroll
    for (int c = 0; c < 3; ++c)
      C[r*3+c] = A[0*3+r]*B[0*3+c] + A[1*3+r]*B[1*3+c] + A[2*3+r]*B[2*3+c];
}
__device__ __forceinline__ void mm3_Bt(const float* A, const float* B, float* C) { // A * B^T
#pragma unroll
  for (int r = 0; r < 3; ++r)
#pragma unroll
    for (int c = 0; c < 3; ++c)
      C[r*3+c] = A[r*3+0]*B[c*3+0] + A[r*3+1]*B[c*3+1] + A[r*3+2]*B[c*3+2];
}
__device__ __forceinline__ void inv3(const float* a, float* o) {
  float c00 = a[4]*a[8] - a[5]*a[7];
  float c01 = a[5]*a[6] - a[3]*a[8];
  float c02 = a[3]*a[7] - a[4]*a[6];
  float det = a[0]*c00 + a[1]*c01 + a[2]*c02;
  float id  = 1.0f / det;
  o[0] = c00*id;                  o[1] = (a[2]*a[7]-a[1]*a[8])*id; o[2] = (a[1]*a[5]-a[2]*a[4])*id;
  o[3] = c01*id;                  o[4] = (a[0]*a[8]-a[2]*a[6])*id; o[5] = (a[2]*a[3]-a[0]*a[5])*id;
  o[6] = c02*id;                  o[7] = (a[1]*a[6]-a[0]*a[7])*id; o[8] = (a[0]*a[4]-a[1]*a[3])*id;
}

// ======================================================================
// Kernel 1: per-node prep.  Lambda_p, Lambda_q, R (Rodrigues), S = R^T Lq R.
// S stored in interleaved row-pair layout:
//   element (row n, col c) -> Sp[((n>>1)*16 + c)*2 + (n&1)]
// so a WMMA B-lane's two K-values are one contiguous float2 (ds_load_b64),
// and the array is linear in K for flat async LDS staging.
// ======================================================================
__global__ void prep_kernel(const float* __restrict__ Sig_p,
                            const float* __restrict__ Sig_q,
                            const float* __restrict__ phi,
                            float* __restrict__ Lp, float* __restrict__ Lq,
                            float* __restrict__ Rm, float* __restrict__ Sp,
                            int N) {
  int n = blockIdx.x * blockDim.x + threadIdx.x;
  if (n >= N) return;
  float A[9], Lpv[9], Lqv[9];
#pragma unroll
  for (int i = 0; i < 9; ++i) A[i] = Sig_p[(size_t)n*9 + i];
  A[0] += EPS_; A[4] += EPS_; A[8] += EPS_;
  inv3(A, Lpv);
#pragma unroll
  for (int i = 0; i < 9; ++i) Lp[(size_t)n*9 + i] = Lpv[i];

#pragma unroll
  for (int i = 0; i < 9; ++i) A[i] = Sig_q[(size_t)n*9 + i];
  A[0] += EPS_; A[4] += EPS_; A[8] += EPS_;
  inv3(A, Lqv);
#pragma unroll
  for (int i = 0; i < 9; ++i) Lq[(size_t)n*9 + i] = Lqv[i];

  // Rodrigues: exp(hat(phi)); hat map matches the reference so(3) generators
  float p0 = phi[(size_t)n*3+0], p1 = phi[(size_t)n*3+1], p2 = phi[(size_t)n*3+2];
  float th2 = p0*p0 + p1*p1 + p2*p2;
  float th = sqrtf(th2);
  float sA, cB;
  if (th < 1e-4f) { sA = 1.0f - th2*(1.0f/6.0f); cB = 0.5f - th2*(1.0f/24.0f); }
  else            { sA = __sinf(th)/th;          cB = (1.0f - __cosf(th))/th2; }
  float R[9];
  R[0] = 1.0f + cB*(p0*p0 - th2);
  R[1] = -sA*p2 + cB*p0*p1;
  R[2] =  sA*p1 + cB*p0*p2;
  R[3] =  sA*p2 + cB*p0*p1;
  R[4] = 1.0f + cB*(p1*p1 - th2);
  R[5] = -sA*p0 + cB*p1*p2;
  R[6] = -sA*p1 + cB*p0*p2;
  R[7] =  sA*p0 + cB*p1*p2;
  R[8] = 1.0f + cB*(p2*p2 - th2);
#pragma unroll
  for (int i = 0; i < 9; ++i) Rm[(size_t)n*9 + i] = R[i];

  float T[9], S[9];
  mm3(Lqv, R, T);     // Lq * R
  mm3_At(R, T, S);    // R^T * Lq * R
#pragma unroll
  for (int cc = 0; cc < 16; ++cc)
    Sp[(((size_t)(n >> 1) * 16 + cc) * 2) + (n & 1)] = (cc < 9) ? S[cc] : 0.0f;
}

// ======================================================================
// Kernel 2: column sums of beta (coalesced: thread = column).
// ======================================================================
__global__ void colsum_kernel(const float* __restrict__ beta,
                              float* __restrict__ bsum, int N) {
  int c = blockIdx.x * blockDim.x + threadIdx.x;
  if (c >= N) return;
  float acc = 0.0f;
  for (int r = 0; r < N; ++r) acc += beta[(size_t)r*N + c];
  bsum[c] = acc;
}

// ======================================================================
// Kernel 3: Lambda_o via fused online-softmax moment reduction over V.
// One 256-thread block per node; LDS merge-reduction of (max, sum, 9 moments).
// ======================================================================
__global__ void lambda_o_kernel(const float* __restrict__ mu,
                                const float* __restrict__ W,
                                float* __restrict__ Lo, int V) {
  const int n = blockIdx.x;
  const int tid = threadIdx.x;
  const float m0 = mu[(size_t)n*3+0], m1 = mu[(size_t)n*3+1], m2 = mu[(size_t)n*3+2];

  float mx = -1e30f, s = 0.0f;
  float a[9];
#pragma unroll
  for (int i = 0; i < 9; ++i) a[i] = 0.0f;

  for (int v = tid; v < V; v += 256) {
    float w0 = W[(size_t)v*3+0], w1 = W[(size_t)v*3+1], w2 = W[(size_t)v*3+2];
    float l = m0*w0 + m1*w1 + m2*w2;       // TAU = 1
    float nm = fmaxf(mx, l);
    float sc = __expf(mx - nm);
    float e  = __expf(l - nm);
    mx = nm;
    s = s*sc + e;
    a[0] = a[0]*sc + e*w0;    a[1] = a[1]*sc + e*w1;    a[2] = a[2]*sc + e*w2;
    a[3] = a[3]*sc + e*w0*w0; a[4] = a[4]*sc + e*w0*w1; a[5] = a[5]*sc + e*w0*w2;
    a[6] = a[6]*sc + e*w1*w1; a[7] = a[7]*sc + e*w1*w2; a[8] = a[8]*sc + e*w2*w2;
  }

  __shared__ float shm[256], shs[256], sha[9][256];
  shm[tid] = mx; shs[tid] = s;
#pragma unroll
  for (int i = 0; i < 9; ++i) sha[i][tid] = a[i];

  for (int st = 128; st > 0; st >>= 1) {
    __syncthreads();
    if (tid < st) {
      float mb = shm[tid + st];
      float nm = fmaxf(shm[tid], mb);
      float f1 = __expf(shm[tid] - nm);
      float f2 = __expf(mb - nm);
      shm[tid] = nm;
      shs[tid] = shs[tid]*f1 + shs[tid + st]*f2;
#pragma unroll
      for (int i = 0; i < 9; ++i) sha[i][tid] = sha[i][tid]*f1 + sha[i][tid + st]*f2;
    }
  }
  __syncthreads();
  if (tid == 0) {
    float inv_s = 1.0f / shs[0];
    float e0 = sha[0][0]*inv_s, e1 = sha[1][0]*inv_s, e2 = sha[2][0]*inv_s;
    float L00 = sha[3][0]*inv_s - e0*e0 + EPS_;
    float L01 = sha[4][0]*inv_s - e0*e1;
    float L02 = sha[5][0]*inv_s - e0*e2;
    float L11 = sha[6][0]*inv_s - e1*e1 + EPS_;
    float L12 = sha[7][0]*inv_s - e1*e2;
    float L22 = sha[8][0]*inv_s - e2*e2 + EPS_;
    float* o = Lo + (size_t)n*9;
    o[0]=L00; o[1]=L01; o[2]=L02; o[3]=L01; o[4]=L11; o[5]=L12; o[6]=L02; o[7]=L12; o[8]=L22;
  }
}

// ======================================================================
// Kernel 4: G = beta @ S via V_WMMA_F32_16X16X4_F32, with the B panel
// double-buffered through LDS using CDNA5 async global->LDS copies
// (global_load_async_to_lds_b128 / s_wait_asynccnt).
// Block = 8 waves; wave w owns 16-row tile blockIdx.x*8 + w.
// Per chunk (256 K-rows, 16KB): each wave issues 4 async b128 ops;
// s_wait_asynccnt 0x4 overlaps chunk c+1's copy with chunk c's WMMAs.
// ======================================================================
#define CHUNK_ROWS 256
#define CHUNK_BYTES (CHUNK_ROWS * 16 * 4)   // 16 KB
#define ASYNC_OPS 4                          // per thread per chunk (b128 each)

__global__ void __launch_bounds__(256)
beta_gemm_wmma(const float* __restrict__ beta,
               const float* __restrict__ Sp,   // interleaved pair layout, linear in K
               float* __restrict__ Minc, int N) {
  __shared__ float sS[2][CHUNK_ROWS * 16];

  const int t    = threadIdx.x;
  const int wave = t >> 5;
  const int lane = t & 31;
  const int half = lane >> 4;      // 0: K+{0,1}, 1: K+{2,3}
  const int l15  = lane & 15;
  const int tile = blockIdx.x * 8 + wave;
  const float* __restrict__ arow = beta + (size_t)(tile*16 + l15) * (size_t)N;
  const char*  __restrict__ Spb  = (const char*)Sp;
  const int NC = N / CHUNK_ROWS;

  // prologue: issue async fill of chunk 0 into buffer 0
  {
    unsigned lds0 = (unsigned)(size_t)(&sS[0][0]);
#pragma unroll
    for (int r = 0; r < ASYNC_OPS; ++r) {
      unsigned loff = lds0 + (unsigned)(t + r*256) * 16u;
      unsigned long long ga = (unsigned long long)(Spb + (size_t)(t + r*256) * 16u);
      asm volatile("global_load_async_to_lds_b128 %0, %1, off"
                   :: "v"(loff), "v"(ga) : "memory");
    }
  }

  v8f c = {};
  for (int ch = 0; ch < NC; ++ch) {
    if (ch + 1 < NC) {
      // issue next chunk's copy before waiting on the current one
      unsigned lds0 = (unsigned)(size_t)(&sS[(ch + 1) & 1][0]);
      const char* gbase = Spb + (size_t)(ch + 1) * CHUNK_BYTES;
#pragma unroll
      for (int r = 0; r < ASYNC_OPS; ++r) {
        unsigned loff = lds0 + (unsigned)(t + r*256) * 16u;
        unsigned long long ga = (unsigned long long)(gbase + (size_t)(t + r*256) * 16u);
        asm volatile("global_load_async_to_lds_b128 %0, %1, off"
                     :: "v"(loff), "v"(ga) : "memory");
      }
      // async loads complete in order: <=4 pending means chunk ch has landed
      asm volatile("s_wait_asynccnt 0x4" ::: "memory");
    } else {
      asm volatile("s_wait_asynccnt 0x0" ::: "memory");
    }
    __syncthreads();   // publish chunk ch across all 8 waves

    const float* __restrict__ bbuf = &sS[ch & 1][0];
    const int kbase = ch * CHUNK_ROWS;
#pragma unroll 4
    for (int kk = 0; kk < CHUNK_ROWS; kk += 4) {
      const int ka = kbase + kk + 2*half;
      v2f a = *(const v2f*)(arow + ka);                       // global_load_b64 (beta stream)
      const int pair = (kk >> 1) + half;                      // interleaved pair index
      v2f b = *(const v2f*)(bbuf + ((size_t)pair*16 + l15)*2); // single ds_load_b64
      c = __builtin_amdgcn_wmma_f32_16x16x4_f32(
          /*neg_a=*/false, a, /*neg_b=*/false, b,
          /*c_mod=*/(short)0, c, /*reuse_a=*/false, /*reuse_b=*/false);
    }
    __syncthreads();   // done reading buf[ch&1] before it is refilled
  }

  if (l15 < 9) {
#pragma unroll
    for (int r = 0; r < 8; ++r) {
      int row = tile*16 + r + 8*half;
      Minc[(size_t)row*9 + l15] = c[r];
    }
  }
}

// ======================================================================
// Kernel 5: finalize.  M = Lp + Lo + R G R^T + colsum*Lq; sym; NaN->I; +1e-4 I.
// Analytic 3x3 eig floor (no-op fast path -> adjugate inverse), faithful
// eigen-reconstruction fallback if the floor binds.
// ======================================================================
__global__ void finalize_kernel(const float* __restrict__ Lp, const float* __restrict__ Lq,
                                const float* __restrict__ Rm, const float* __restrict__ Lo,
                                const float* __restrict__ bsum, const float* __restrict__ Minc,
                                float* __restrict__ out, int N) {
  int n = blockIdx.x * blockDim.x + threadIdx.x;
  if (n >= N) return;

  float R[9], G[9], T[9], F[9], M[9];
#pragma unroll
  for (int i = 0; i < 9; ++i) { R[i] = Rm[(size_t)n*9+i]; G[i] = Minc[(size_t)n*9+i]; }
  mm3(R, G, T);
  mm3_Bt(T, R, F);                       // R G R^T
  float bs = bsum[n];
#pragma unroll
  for (int i = 0; i < 9; ++i)
    M[i] = Lp[(size_t)n*9+i] + Lo[(size_t)n*9+i] + F[i] + bs * Lq[(size_t)n*9+i];
  // symmetrize
  float t;
  t = 0.5f*(M[1]+M[3]); M[1]=t; M[3]=t;
  t = 0.5f*(M[2]+M[6]); M[2]=t; M[6]=t;
  t = 0.5f*(M[5]+M[7]); M[5]=t; M[7]=t;
  // non-finite -> identity entries
#pragma unroll
  for (int i = 0; i < 9; ++i)
    if (!__builtin_isfinite(M[i])) M[i] = ((i & 3) == 0) ? 1.0f : 0.0f;
  M[0] += MINEIG_; M[4] += MINEIG_; M[8] += MINEIG_;

  // analytic eigenvalues of symmetric 3x3
  float q  = (M[0] + M[4] + M[8]) * (1.0f/3.0f);
  float pp = M[1]*M[1] + M[2]*M[2] + M[5]*M[5];
  float p2 = (M[0]-q)*(M[0]-q) + (M[4]-q)*(M[4]-q) + (M[8]-q)*(M[8]-q) + 2.0f*pp;
  float p  = sqrtf(p2 * (1.0f/6.0f));
  float w0, w1, w2;
  if (p < 1e-12f) { w0 = w1 = w2 = q; }
  else {
    float ip = 1.0f / p;
    float B0 = (M[0]-q)*ip, B4 = (M[4]-q)*ip, B8 = (M[8]-q)*ip;
    float B1 = M[1]*ip, B2 = M[2]*ip, B5 = M[5]*ip;
    float detB = B0*(B4*B8 - B5*B5) - B1*(B1*B8 - B5*B2) + B2*(B1*B5 - B4*B2);
    float r = fminf(fmaxf(0.5f*detB, -1.0f), 1.0f);
    float ph = acosf(r) * (1.0f/3.0f);
    w0 = q + 2.0f*p*__cosf(ph);
    w2 = q + 2.0f*p*__cosf(ph + 2.09439510239f);
    w1 = 3.0f*q - w0 - w2;
  }
  float wmin = fminf(w0, fminf(w1, w2));

  float* Opd  = out + (size_t)n*9;
  float* Oinv = out + (size_t)N*9 + (size_t)n*9;

  if (wmin >= MINEIG_) {
    // clip is a no-op: M_pd = M exactly, M_inv = adjugate inverse
    float Mi[9];
    inv3(M, Mi);
#pragma unroll
    for (int i = 0; i < 9; ++i) { Opd[i] = M[i]; Oinv[i] = Mi[i]; }
  } else {
    // eigenvectors via Cayley-Hamilton products
    float u0[3], u1[3], u2[3];
    {
      float A1[9], A2[9], C[9];
#pragma unroll
      for (int i = 0; i < 9; ++i) { A1[i] = M[i]; A2[i] = M[i]; }
      A1[0]-=w1; A1[4]-=w1; A1[8]-=w1;  A2[0]-=w2; A2[4]-=w2; A2[8]-=w2;
      mm3(A1, A2, C);
      float best = -1.0f; int bc = 0;
#pragma unroll
      for (int cc = 0; cc < 3; ++cc) {
        float n2 = C[cc]*C[cc] + C[3+cc]*C[3+cc] + C[6+cc]*C[6+cc];
        if (n2 > best) { best = n2; bc = cc; }
      }
      float iv = rsqrtf(fmaxf(best, 1e-30f));
      u0[0]=C[bc]*iv; u0[1]=C[3+bc]*iv; u0[2]=C[6+bc]*iv;
    }
    {
      float A1[9], A2[9], C[9];
#pragma unroll
      for (int i = 0; i < 9; ++i) { A1[i] = M[i]; A2[i] = M[i]; }
      A1[0]-=w0; A1[4]-=w0; A1[8]-=w0;  A2[0]-=w1; A2[4]-=w1; A2[8]-=w1;
      mm3(A1, A2, C);
      float best = -1.0f; int bc = 0;
#pragma unroll
      for (int cc = 0; cc < 3; ++cc) {
        float n2 = C[cc]*C[cc] + C[3+cc]*C[3+cc] + C[6+cc]*C[6+cc];
        if (n2 > best) { best = n2; bc = cc; }
      }
      float iv = rsqrtf(fmaxf(best, 1e-30f));
      u2[0]=C[bc]*iv; u2[1]=C[3+bc]*iv; u2[2]=C[6+bc]*iv;
    }
    u1[0] = u2[1]*u0[2] - u2[2]*u0[1];
    u1[1] = u2[2]*u0[0] - u2[0]*u0[2];
    u1[2] = u2[0]*u0[1] - u2[1]*u0[0];
    float iv = rsqrtf(fmaxf(u1[0]*u1[0]+u1[1]*u1[1]+u1[2]*u1[2], 1e-30f));
    u1[0]*=iv; u1[1]*=iv; u1[2]*=iv;

    float c0 = fmaxf(w0, MINEIG_), c1 = fmaxf(w1, MINEIG_), c2 = fmaxf(w2, MINEIG_);
#pragma unroll
    for (int r = 0; r < 3; ++r)
#pragma unroll
      for (int cc = 0; cc < 3; ++cc) {
        Opd[r*3+cc]  = c0*u0[r]*u0[cc] + c1*u1[r]*u1[cc] + c2*u2[r]*u2[cc];
        Oinv[r*3+cc] = u0[r]*u0[cc]/c0 + u1[r]*u1[cc]/c1 + u2[r]*u2[cc]/c2;
      }
  }
}

// ======================================================================
extern "C" void kernel_launch(void* const* d_in, const int* in_sizes, int n_in,
                              void* d_out, int out_size, void* d_ws, size_t ws_size,
                              hipStream_t stream) {
  const float* Sigma_prior = (const float*)d_in[0];
  const float* Sigma_q     = (const float*)d_in[1];
  const float* phi         = (const float*)d_in[2];
  const float* beta        = (const float*)d_in[3];
  const float* mu          = (const float*)d_in[4];
  const float* W_out       = (const float*)d_in[5];
  // d_in[6] = generators (constant so(3) basis, hardcoded in prep_kernel)

  const int N = in_sizes[4] / 3;   // mu: (B=1, N, K=3)
  const int V = in_sizes[5] / 3;   // W_out: (V, K=3)
  float* out = (float*)d_out;      // [M_pd (N*9) | M_inv (N*9)]

  // workspace layout (floats): Lp, Lq, R, Sp(16-wide pairs), Lo, bsum, Minc
  float* ws   = (float*)d_ws;
  float* Lp   = ws;
  float* Lq   = Lp + (size_t)N*9;
  float* Rm   = Lq + (size_t)N*9;
  float* Sp   = Rm + (size_t)N*9;
  float* Lo   = Sp + (size_t)N*16;
  float* bsum = Lo + (size_t)N*9;
  float* Minc = bsum + (size_t)N;

  const int TPB = 256;
  const int nblk = (N + TPB - 1) / TPB;

  prep_kernel<<<nblk, TPB, 0, stream>>>(Sigma_prior, Sigma_q, phi, Lp, Lq, Rm, Sp, N);
  colsum_kernel<<<nblk, TPB, 0, stream>>>(beta, bsum, N);
  lambda_o_kernel<<<N, TPB, 0, stream>>>(mu, W_out, Lo, V);
  beta_gemm_wmma<<<N / 128, 256, 0, stream>>>(beta, Sp, Minc, N);
  finalize_kernel<<<nblk, TPB, 0, stream>>>(Lp, Lq, Rm, Lo, bsum, Minc, out, N);
}